// GCN_22273700397686
// MI455X (gfx1250) — compile-verified
//
#include <hip/hip_runtime.h>
#include <hip/hip_bf16.h>

typedef float v2f __attribute__((ext_vector_type(2)));
typedef float v8f __attribute__((ext_vector_type(8)));

#define D 32

// ---------------- degree / normalization ----------------

__global__ void k_init_deg(float* deg, int n) {
    int i = blockIdx.x * blockDim.x + threadIdx.x;
    if (i < n) deg[i] = 1.0f;                 // self-loop contribution
}

__global__ void k_count_deg(const int* __restrict__ dst, float* deg, int E) {
    int e = blockIdx.x * blockDim.x + threadIdx.x;
    if (e < E) unsafeAtomicAdd(&deg[dst[e]], 1.0f);   // global_atomic_add_f32
}

__global__ void k_rsqrt(float* deg, int n) {
    int i = blockIdx.x * blockDim.x + threadIdx.x;
    if (i < n) deg[i] = rsqrtf(deg[i]);       // deg >= 1 always (self-loops)
}

// ---------------- H = X @ W via V_WMMA_F32_16X16X4_F32 ----------------
// Also writes AGG = H * dis^2 (self-loop init of the aggregation buffer).
// AGG may alias X: each wave reads its 16 rows completely before storing them,
// and rows are private to one wave.
__global__ void k_gemm_wmma(const float* X, const float* __restrict__ W,
                            const float* __restrict__ dis,
                            float* __restrict__ H, float* AGG, int n) {
    int wave = threadIdx.x >> 5;
    int lane = threadIdx.x & 31;
    int tile = blockIdx.x * (blockDim.x >> 5) + wave;
    if (tile * 16 >= n) return;               // wave-uniform: EXEC stays all-1s

    int half = lane >> 4;                     // 0: lanes 0-15 (K=0,1), 1: lanes 16-31 (K=2,3)
    int m    = lane & 15;
    const float* xrow = X + (size_t)(tile * 16 + m) * D;

    v8f acc0 = {};                            // cols 0..15
    v8f acc1 = {};                            // cols 16..31
#pragma unroll
    for (int kk = 0; kk < 8; ++kk) {          // K = 32 in steps of 4
        int k0 = 4 * kk + 2 * half;
        v2f a, b0, b1;
        a.x  = xrow[k0];            a.y  = xrow[k0 + 1];
        b0.x = W[k0 * D + m];       b0.y = W[(k0 + 1) * D + m];
        b1.x = W[k0 * D + 16 + m];  b1.y = W[(k0 + 1) * D + 16 + m];
        acc0 = __builtin_amdgcn_wmma_f32_16x16x4_f32(false, a, false, b0,
                                                     (short)0, acc0, false, false);
        acc1 = __builtin_amdgcn_wmma_f32_16x16x4_f32(false, a, false, b1,
                                                     (short)0, acc1, false, false);
    }

#pragma unroll
    for (int r = 0; r < 8; ++r) {             // D layout: VGPR r -> M = r + 8*half, N = m
        int row = tile * 16 + r + 8 * half;
        float s  = dis[row]; s = s * s;
        float h0 = acc0[r], h1 = acc1[r];
        H[(size_t)row * D + m]        = h0;
        H[(size_t)row * D + 16 + m]   = h1;
        AGG[(size_t)row * D + m]      = h0 * s;
        AGG[(size_t)row * D + 16 + m] = h1 * s;
    }
}

// ---------------- edge aggregation: AGG[dst] += H[src] * dis[src]*dis[dst] ----------------
// 8 threads per edge, float4 each: the 8 lanes cover one 128B row -> coalesced
// gather; scatter uses hardware fp32 atomics into L2 (whole array is L2-resident).
__global__ void k_edge(const float* __restrict__ H, const float* __restrict__ dis,
                       const int* __restrict__ src, const int* __restrict__ dst,
                       float* AGG, int E) {
    int tid = blockIdx.x * blockDim.x + threadIdx.x;
    int e = tid >> 3;
    if (e >= E) return;
    int c = (tid & 7) << 2;
    int s = src[e], d = dst[e];
    float nrm = dis[s] * dis[d];
    const float4* hp = (const float4*)(H + (size_t)s * D + c);
    float4 v = *hp;
    float* ap = AGG + (size_t)d * D + c;
    unsafeAtomicAdd(ap + 0, v.x * nrm);
    unsafeAtomicAdd(ap + 1, v.y * nrm);
    unsafeAtomicAdd(ap + 2, v.z * nrm);
    unsafeAtomicAdd(ap + 3, v.w * nrm);
}

// ---------------- bias (+ optional ReLU) ----------------
__global__ void k_bias_act(float* A, const float* __restrict__ b, int total, int relu) {
    int i = blockIdx.x * blockDim.x + threadIdx.x;
    if (i >= total) return;
    float v = A[i] + b[i & (D - 1)];
    if (relu) v = fmaxf(v, 0.0f);
    A[i] = v;
}

// ---------------- readout: out = H @ Wlin + blin  (32 -> 2) ----------------
__global__ void k_final(const float* __restrict__ H, const float* __restrict__ Wl,
                        const float* __restrict__ bl, float* __restrict__ out, int n) {
    int i = blockIdx.x * blockDim.x + threadIdx.x;
    if (i >= n) return;
    float a0 = bl[0], a1 = bl[1];
    const float4* hp = (const float4*)(H + (size_t)i * D);
#pragma unroll
    for (int q = 0; q < 8; ++q) {
        float4 v = hp[q];
        a0 += v.x * Wl[(4*q+0)*2]   + v.y * Wl[(4*q+1)*2]
            + v.z * Wl[(4*q+2)*2]   + v.w * Wl[(4*q+3)*2];
        a1 += v.x * Wl[(4*q+0)*2+1] + v.y * Wl[(4*q+1)*2+1]
            + v.z * Wl[(4*q+2)*2+1] + v.w * Wl[(4*q+3)*2+1];
    }
    out[(size_t)i * 2 + 0] = a0;
    out[(size_t)i * 2 + 1] = a1;
}

// ---------------- driver ----------------
extern "C" void kernel_launch(void* const* d_in, const int* in_sizes, int n_in,
                              void* d_out, int out_size, void* d_ws, size_t ws_size,
                              hipStream_t stream) {
    const float* x    = (const float*)d_in[0];
    const int*   ei   = (const int*)  d_in[1];
    const float* W1   = (const float*)d_in[2];
    const float* b1   = (const float*)d_in[3];
    const float* W2   = (const float*)d_in[4];
    const float* b2   = (const float*)d_in[5];
    const float* W3   = (const float*)d_in[6];
    const float* b3   = (const float*)d_in[7];
    const float* Wlin = (const float*)d_in[8];
    const float* blin = (const float*)d_in[9];

    int n = in_sizes[0] / D;
    int E = in_sizes[1] / 2;
    const int* src = ei;
    const int* dst = ei + E;

    float* dis = (float*)d_ws;                // n floats (deg -> dis in place)
    float* B0  = dis + n;                     // n*D floats (H buffer)
    float* B1  = B0 + (size_t)n * D;          // n*D floats (AGG / cur buffer)
    float* out = (float*)d_out;

    const int TB = 256;
    int gN  = (n + TB - 1) / TB;
    int gE  = (E + TB - 1) / TB;
    int gE8 = (E * 8 + TB - 1) / TB;
    int gEl = (n * D + TB - 1) / TB;
    int tiles = (n + 15) / 16;
    int gGemm = (tiles + 7) / 8;              // 8 waves (tiles) per 256-thread block

    // normalization: deg includes self-loop, dis = deg^{-1/2}
    k_init_deg <<<gN, TB, 0, stream>>>(dis, n);
    k_count_deg<<<gE, TB, 0, stream>>>(dst, dis, E);
    k_rsqrt    <<<gN, TB, 0, stream>>>(dis, n);

    // layer 1: cur = x
    k_gemm_wmma<<<gGemm, 256, 0, stream>>>(x,  W1, dis, B0, B1, n);
    k_edge     <<<gE8,  TB,  0, stream>>>(B0, dis, src, dst, B1, E);
    k_bias_act <<<gEl,  TB,  0, stream>>>(B1, b1, n * D, 1);

    // layer 2: cur = B1 (AGG aliases cur inside gemm — safe, see kernel comment)
    k_gemm_wmma<<<gGemm, 256, 0, stream>>>(B1, W2, dis, B0, B1, n);
    k_edge     <<<gE8,  TB,  0, stream>>>(B0, dis, src, dst, B1, E);
    k_bias_act <<<gEl,  TB,  0, stream>>>(B1, b2, n * D, 1);

    // layer 3 (no ReLU)
    k_gemm_wmma<<<gGemm, 256, 0, stream>>>(B1, W3, dis, B0, B1, n);
    k_edge     <<<gE8,  TB,  0, stream>>>(B0, dis, src, dst, B1, E);
    k_bias_act <<<gEl,  TB,  0, stream>>>(B1, b3, n * D, 0);

    // readout
    k_final<<<gN, TB, 0, stream>>>(B1, Wlin, blin, out, n);
}